// PtsManipulator_11201274708088
// MI455X (gfx1250) — compile-verified
//
#include <hip/hip_runtime.h>
#include <math.h>
#include <stdint.h>

// ---------------------------------------------------------------------------
// Problem constants (match reference)
// ---------------------------------------------------------------------------
#define HH    256
#define WW    256
#define CC    3
#define NPTS  (HH * WW)        // 65536 points / pixels per batch
#define BS    2
#define KPPP  8                // top-K per pixel
#define EPSV  0.01f
#define RAD2  2.25f            // RADIUS_PX^2

// Binning grid: pixel p gathers points with floor(x) in [p-2, p+1]
// (|p - x| <= 1.5  =>  p - floor(x) in {-1,0,1,2}).  Pad grid by 2 each side.
#define BPAD  2
#define BW    (WW + 2 * BPAD)  // 260
#define BH    (HH + 2 * BPAD)  // 260
#define CELLS (BW * BH)        // 67600 cells per batch

typedef float v2f __attribute__((ext_vector_type(2)));
typedef float v8f __attribute__((ext_vector_type(8)));

// ---------------------------------------------------------------------------
// Kernel 1: fold K * RTinv_cam2 * RT_cam1 * K_inv into one 4x4 per batch.
// ---------------------------------------------------------------------------
__global__ void k_matchain(const float* __restrict__ K, const float* __restrict__ Kinv,
                           const float* __restrict__ RT1, const float* __restrict__ RTi2,
                           float* __restrict__ M) {
  int t = threadIdx.x;
  if (t >= BS * 16) return;
  int b = t >> 4, e = t & 15;
  int r = e >> 2, c = e & 3;
  const float* Kb = K    + b * 16;
  const float* Ib = Kinv + b * 16;
  const float* R1 = RT1  + b * 16;
  const float* R2 = RTi2 + b * 16;
  float acc = 0.0f;
  for (int j = 0; j < 4; ++j)
    for (int k = 0; k < 4; ++k)
      for (int l = 0; l < 4; ++l)
        acc += Kb[r * 4 + j] * R2[j * 4 + k] * R1[k * 4 + l] * Ib[l * 4 + c];
  M[b * 16 + e] = acc;
}

// ---------------------------------------------------------------------------
// Kernel 2: projection via V_WMMA_F32_16X16X4_F32 (one WMMA per 16 points).
//   A (16x4): combined-matrix rows replicated 4x; VGPR0=K(0|2), VGPR1=K(1|3).
//   B (4x16): column n = point (wave*16 + lane%16), vector [X*d, Y*d, d, 1].
//   D rows 0..2 (rows 8..10 duplicate them for the upper half-wave) = x',y',z'.
// Writes packed {x_pix, y_pix, z, alpha} per point.
// ---------------------------------------------------------------------------
__global__ void k_project(const float* __restrict__ pred, const float* __restrict__ alphas,
                          const float* __restrict__ M, float4* __restrict__ pts4) {
  const int lane = threadIdx.x & 31;
  const int wave = (blockIdx.x * blockDim.x + threadIdx.x) >> 5;
  const int col  = lane & 15;
  const int p    = wave * 16 + col;     // global point id in [0, BS*NPTS)
  const int b    = p >> 16;             // / NPTS
  const int pid  = p & (NPTS - 1);
  const float X  = (float)(pid & (WW - 1));
  const float Y  = (float)(pid >> 8);
  const float d  = pred[p];
  const float al = alphas[p];

  const float* CM = M + b * 16;
  const int hi   = lane >> 4;           // 0: K pair {0,1}, 1: K pair {2,3}
  const int arow = lane & 3;            // matrix row (replicated 4x over M dim)
  v2f av, bv;
  av.x = CM[arow * 4 + (hi ? 2 : 0)];
  av.y = CM[arow * 4 + (hi ? 3 : 1)];
  bv.x = hi ? d    : X * d;             // K=2 (d)   | K=0 (X*d)
  bv.y = hi ? 1.0f : Y * d;             // K=3 (1)   | K=1 (Y*d)

  v8f cacc = {};
  cacc = __builtin_amdgcn_wmma_f32_16x16x4_f32(
      /*neg_a=*/false, av, /*neg_b=*/false, bv,
      /*c_mod=*/(short)0, cacc, /*reuse_a=*/false, /*reuse_b=*/false);

  float wx = cacc[0], wy = cacc[1], wz = cacc[2];
  bool  msk   = fabsf(wz) < EPSV;
  float zsafe = msk ? EPSV : wz;
  float xp = wx / zsafe;                // == (sx+1)*0.5*(W-1) in the reference
  float yp = wy / zsafe;
  float zo = wz;
  if (msk) { xp = -1.0e9f; yp = -1.0e9f; zo = -10.0f; }  // sampler=-10 path
  if (lane < 16) pts4[p] = make_float4(xp, yp, zo, al);
}

// ---------------------------------------------------------------------------
// Binning helper.
// ---------------------------------------------------------------------------
__device__ __forceinline__ int bin_cell(float x, float y, int b, bool& ok) {
  int icx = (int)floorf(x);
  int icy = (int)floorf(y);
  ok = !(icx < -BPAD || icx > WW + BPAD - 1 || icy < -BPAD || icy > HH + BPAD - 1);
  return b * CELLS + (icy + BPAD) * BW + (icx + BPAD);
}

// ---------------------------------------------------------------------------
// Kernel 3: count points per bin cell.
// ---------------------------------------------------------------------------
__global__ void k_count(const float4* __restrict__ pts4, int* __restrict__ counts) {
  int g = blockIdx.x * blockDim.x + threadIdx.x;
  if (g >= BS * NPTS) return;
  float4 q = pts4[g];
  bool ok;
  int cell = bin_cell(q.x, q.y, g >> 16, ok);
  if (ok) atomicAdd(&counts[cell], 1);
}

// ---------------------------------------------------------------------------
// Kernel 4: exclusive scan over all BS*CELLS counts (single workgroup).
// Chunk loads go global -> LDS via the gfx1250 async-copy path
// (global_load_async_to_lds_b32, tracked by ASYNCcnt), then Hillis-Steele
// scan per 1024-chunk with carried prefix.  Writes starts + cursors.
// ---------------------------------------------------------------------------
__global__ void k_scan(const int* __restrict__ counts, int* __restrict__ starts,
                       int* __restrict__ cursors) {
  __shared__ int sh[1024];
  __shared__ int carry_s;
  const int n = BS * CELLS;
  if (threadIdx.x == 0) carry_s = 0;
  __syncthreads();
  for (int base = 0; base < n; base += 1024) {
    int i = base + threadIdx.x;
    if (i < n) {
      // Async copy counts[i] -> sh[threadIdx.x].  Generic LDS pointers keep
      // the LDS byte offset in their low 32 bits (aperture in high bits).
      unsigned lds_addr = (unsigned)(uintptr_t)&sh[threadIdx.x];
      const int* gp = counts + i;
      asm volatile("global_load_async_to_lds_b32 %0, %1, off"
                   :: "v"(lds_addr), "v"(gp) : "memory");
    }
    asm volatile("s_wait_asynccnt 0x0" ::: "memory");
    __syncthreads();
    int v = (i < n) ? sh[threadIdx.x] : 0;
    __syncthreads();
    sh[threadIdx.x] = v;                  // normalize tail lanes to zero
    __syncthreads();
    for (int off = 1; off < 1024; off <<= 1) {
      int t = (threadIdx.x >= off) ? sh[threadIdx.x - off] : 0;
      __syncthreads();
      sh[threadIdx.x] += t;
      __syncthreads();
    }
    int incl  = sh[threadIdx.x];
    int carry = carry_s;
    int excl  = carry + incl - v;
    if (i < n) { starts[i] = excl; cursors[i] = excl; }
    __syncthreads();
    if (threadIdx.x == 0) carry_s = carry + sh[1023];
    __syncthreads();
  }
}

// ---------------------------------------------------------------------------
// Kernel 5: scatter packed payload {x,y,z,alpha} + pid into per-cell lists,
// so the composite inner loop streams one b128 + one b32 per candidate.
// ---------------------------------------------------------------------------
__global__ void k_scatter(const float4* __restrict__ pts4, int* __restrict__ cursors,
                          float4* __restrict__ payload, int* __restrict__ plist) {
  int g = blockIdx.x * blockDim.x + threadIdx.x;
  if (g >= BS * NPTS) return;
  float4 q = pts4[g];
  bool ok;
  int cell = bin_cell(q.x, q.y, g >> 16, ok);
  if (!ok) return;
  int pos = atomicAdd(&cursors[cell], 1);
  payload[pos] = q;
  plist[pos]   = g & (NPTS - 1);        // pid within batch
}

// ---------------------------------------------------------------------------
// Kernel 6: per-pixel gather over 4x4 neighbor cells, register top-8 by
// (z asc, pid asc) — pid tie-break makes the result independent of atomic
// scatter order (determinism) and matches the reference lexsort semantics —
// then front-to-back alpha compositing of the 3 feature channels.
// ---------------------------------------------------------------------------
__global__ void k_composite(const float4* __restrict__ payload, const int* __restrict__ plist,
                            const float* __restrict__ src, const int* __restrict__ starts,
                            const int* __restrict__ cursors, float* __restrict__ out) {
  int g = blockIdx.x * blockDim.x + threadIdx.x;
  if (g >= BS * NPTS) return;
  int b  = g >> 16;
  int pp = g & (NPTS - 1);
  int py = pp >> 8, px = pp & (WW - 1);
  float fx = (float)px, fy = (float)py;

  float tz[KPPP]; float ta[KPPP]; int tp[KPPP];
#pragma unroll
  for (int k = 0; k < KPPP; ++k) { tz[k] = 3.4e38f; ta[k] = 0.0f; tp[k] = 0x7fffffff; }

  const int cb = b * CELLS;
  for (int dy = 0; dy < 4; ++dy) {
    int rowb = cb + (py + dy) * BW + px;
    for (int dx = 0; dx < 4; ++dx) {
      int cell = rowb + dx;
      int s = starts[cell];
      int e = cursors[cell];              // starts + count
      if (s < e) __builtin_prefetch(&payload[s], 0, 0);  // global_prefetch_b8
      for (int j = s; j < e; ++j) {
        float4 q  = payload[j];           // one b128, sequential within cell
        float ddx = fx - q.x, ddy = fy - q.y;
        float d2  = ddx * ddx + ddy * ddy;
        if (d2 <= RAD2 && q.z > 0.0f) {
          int   pid = plist[j];
          float dn  = d2 * (4.0f / 9.0f); // d2_pix / RADIUS^2
          dn = fminf(fmaxf(dn, 0.001f), 1.0f);
          float a = (1.0f - sqrtf(dn)) * q.w;
          bool better = (q.z < tz[KPPP - 1]) ||
                        (q.z == tz[KPPP - 1] && pid < tp[KPPP - 1]);
          if (better) {
            tz[KPPP - 1] = q.z; ta[KPPP - 1] = a; tp[KPPP - 1] = pid;
#pragma unroll
            for (int k = KPPP - 1; k > 0; --k) {
              bool sw = (tz[k] < tz[k - 1]) ||
                        (tz[k] == tz[k - 1] && tp[k] < tp[k - 1]);
              if (sw) {
                float t0 = tz[k]; tz[k] = tz[k - 1]; tz[k - 1] = t0;
                float t1 = ta[k]; ta[k] = ta[k - 1]; ta[k - 1] = t1;
                int   t2 = tp[k]; tp[k] = tp[k - 1]; tp[k - 1] = t2;
              }
            }
          }
        }
      }
    }
  }

  float T = 1.0f, o0 = 0.0f, o1 = 0.0f, o2 = 0.0f;
#pragma unroll
  for (int k = 0; k < KPPP; ++k) {
    if (tp[k] < NPTS) {
      float w  = ta[k] * T;
      int   si = b * CC * NPTS + tp[k];
      o0 += w * src[si];
      o1 += w * src[si + NPTS];
      o2 += w * src[si + 2 * NPTS];
      T *= (1.0f - ta[k]);
    }
  }
  int ob = b * CC * NPTS + pp;
  out[ob]            = o0;
  out[ob + NPTS]     = o1;
  out[ob + 2 * NPTS] = o2;
}

// ---------------------------------------------------------------------------
// Launch
// ---------------------------------------------------------------------------
extern "C" void kernel_launch(void* const* d_in, const int* in_sizes, int n_in,
                              void* d_out, int out_size, void* d_ws, size_t ws_size,
                              hipStream_t stream) {
  (void)in_sizes; (void)n_in; (void)out_size; (void)ws_size;
  const float* alphas = (const float*)d_in[0];
  const float* src    = (const float*)d_in[1];
  const float* pred   = (const float*)d_in[2];
  const float* Km     = (const float*)d_in[3];
  const float* Kinv   = (const float*)d_in[4];
  const float* RT1    = (const float*)d_in[5];
  const float* RTi2   = (const float*)d_in[8];
  float* out = (float*)d_out;

  // Workspace carve-out (~6.3 MB total, 256B-aligned slices)
  char* ws = (char*)d_ws;
  size_t off = 0;
  auto carve = [&](size_t bytes) {
    void* p = ws + off;
    off += (bytes + 255) & ~(size_t)255;
    return p;
  };
  float*  M       = (float*)carve(BS * 16 * sizeof(float));
  float4* pts4    = (float4*)carve(BS * NPTS * sizeof(float4));
  float4* payload = (float4*)carve(BS * NPTS * sizeof(float4));
  int*    plist   = (int*)carve(BS * NPTS * sizeof(int));
  int*    counts  = (int*)carve(BS * CELLS * sizeof(int));
  int*    starts  = (int*)carve(BS * CELLS * sizeof(int));
  int*    cursors = (int*)carve(BS * CELLS * sizeof(int));

  hipMemsetAsync(counts, 0, BS * CELLS * sizeof(int), stream);
  k_matchain<<<1, 32, 0, stream>>>(Km, Kinv, RT1, RTi2, M);
  // 256 threads = 8 waves x 16 points = 128 points/block; no tail.
  k_project<<<(BS * NPTS) / 128, 256, 0, stream>>>(pred, alphas, M, pts4);
  k_count<<<(BS * NPTS) / 256, 256, 0, stream>>>(pts4, counts);
  k_scan<<<1, 1024, 0, stream>>>(counts, starts, cursors);
  k_scatter<<<(BS * NPTS) / 256, 256, 0, stream>>>(pts4, cursors, payload, plist);
  k_composite<<<(BS * NPTS) / 256, 256, 0, stream>>>(payload, plist, src,
                                                     starts, cursors, out);
}